// CondGCN_65627100283094
// MI455X (gfx1250) — compile-verified
//
#include <hip/hip_runtime.h>

typedef __attribute__((ext_vector_type(2))) float v2f;
typedef __attribute__((ext_vector_type(8))) float v8f;

#define NXN 100000
#define NCN 100000
#define EXXE 1600000
#define ECXE 800000

static __device__ __forceinline__ v8f wmma4(v2f a, v2f b, v8f c) {
  // V_WMMA_F32_16X16X4_F32: D = A(16x4 f32) * B(4x16 f32) + C(16x16 f32)
  return __builtin_amdgcn_wmma_f32_16x16x4_f32(false, a, false, b, (short)0, c,
                                               false, false);
}

// Stage a K x 64 row-major weight matrix into LDS pre-swizzled into WMMA
// B-fragment order: frag (kc, nt), lane ln, element j (j = 0,1 contiguous):
//   dst[((kc*4 + nt)*32 + ln)*2 + j] = W[(kc*4 + (ln>>4)*2 + j)*64 + nt*16 + (ln&15)]
// so each lane fetches its whole fragment with ONE aligned ds_load_b64.
static __device__ __forceinline__ void stage_swizzled(float* __restrict__ dst,
                                                      const float* __restrict__ W,
                                                      int totalFloats /* K*64 */,
                                                      int tid) {
  for (int f = tid; f < totalFloats; f += 128) {
    const int j = f & 1;
    const int ln = (f >> 1) & 31;
    const int nt = (f >> 6) & 3;
    const int kc = f >> 8;
    const int row = kc * 4 + (ln >> 4) * 2 + j;
    const int col = nt * 16 + (ln & 15);
    dst[f] = W[row * 64 + col];
  }
}

static __device__ __forceinline__ v2f bfrag(const float* __restrict__ sW,
                                            int kc, int nt, int lane) {
  return *reinterpret_cast<const v2f*>(&sW[kc * 256 + nt * 64 + lane * 2]);
}

__global__ void init_ws_kernel(int* __restrict__ last_xx, int* __restrict__ cnt_xx,
                               int* __restrict__ last_cx, int* __restrict__ cnt_cx) {
  int i = blockIdx.x * blockDim.x + threadIdx.x;
  if (i < NXN) {
    last_xx[i] = -1;
    cnt_xx[i] = 0;
    last_cx[i] = -1;
    cnt_cx[i] = 0;
  }
}

__global__ void scatter_edges_kernel(const int* __restrict__ dst, int E,
                                     int* __restrict__ last, int* __restrict__ cnt) {
  int e = blockIdx.x * blockDim.x + threadIdx.x;
  if (e < E) {
    int d = dst[e];
    atomicMax(&last[d], e);
    atomicAdd(&cnt[d], 1);
  }
}

// One wave32 per 16-node tile; 4 waves (4 tiles) per 128-thread block.
__global__ void __launch_bounds__(128)
x_forward_kernel(const float* __restrict__ x, const float* __restrict__ cfeat,
                 const int* __restrict__ exx_src, const int* __restrict__ ecx_src,
                 const int* __restrict__ last_xx, const int* __restrict__ cnt_xx,
                 const int* __restrict__ last_cx, const int* __restrict__ cnt_cx,
                 const float* __restrict__ Wxx, const float* __restrict__ bxx,
                 const float* __restrict__ Wxex, const float* __restrict__ bxex,
                 const float* __restrict__ Wcx, const float* __restrict__ bcx,
                 const float* __restrict__ Wpool, const float* __restrict__ bpool,
                 float* __restrict__ out) {
  __shared__ __align__(16) float sWxx[64 * 64];    // swizzled fragments
  __shared__ __align__(16) float sWxex[64 * 64];
  __shared__ __align__(16) float sWcx[32 * 64];
  __shared__ __align__(16) float sWpool[64 * 64];
  __shared__ __align__(16) float sBias[4 * 64];    // bxx | bxex | bcx | bpool
  __shared__ __align__(16) float sStage[4][16 * 64];  // per-wave agg transpose

  const int tid = threadIdx.x;
  stage_swizzled(sWxx, Wxx, 64 * 64, tid);
  stage_swizzled(sWxex, Wxex, 64 * 64, tid);
  stage_swizzled(sWpool, Wpool, 64 * 64, tid);
  stage_swizzled(sWcx, Wcx, 32 * 64, tid);
  if (tid < 64) {
    sBias[tid] = bxx[tid];
    sBias[64 + tid] = bxex[tid];
    sBias[128 + tid] = bcx[tid];
    sBias[192 + tid] = bpool[tid];
  }
  __syncthreads();

  const int wave = tid >> 5;
  const int lane = tid & 31;
  const int hb = lane >> 4;   // which half-wave
  const int mr = lane & 15;   // A row index / B-C-D column index within tile
  const int kb = hb * 2;      // K offset within 4-chunk for this half

  const int tile = blockIdx.x * 4 + wave;
  const bool tvalid = (tile * 16) < NXN;
  const int tbase = tvalid ? tile * 16 : 0;  // safe base for inactive tail waves

  // Per-lane gather setup for A-fragment row mr of this tile.
  const int node = tbase + mr;
  const int lxx = last_xx[node];
  const int lcx = last_cx[node];
  const int sxx = exx_src[lxx < 0 ? 0 : lxx];
  const int scx = ecx_src[lcx < 0 ? 0 : lcx];

  const float* rowSelf = x + (size_t)node * 64;
  const float* rowGx = x + (size_t)sxx * 64;
  const float* rowGc = cfeat + (size_t)scx * 32;

  // Per-lane dup-dst counts for the C-layout rows this lane owns (hb*8 + r):
  // 8 contiguous ints -> two int4 loads.
  float cntx[8], cntc[8];
  {
    const int4* px = reinterpret_cast<const int4*>(cnt_xx + tbase + hb * 8);
    const int4* pc = reinterpret_cast<const int4*>(cnt_cx + tbase + hb * 8);
    int4 x0 = px[0], x1 = px[1], c0 = pc[0], c1 = pc[1];
    cntx[0] = (float)x0.x; cntx[1] = (float)x0.y; cntx[2] = (float)x0.z; cntx[3] = (float)x0.w;
    cntx[4] = (float)x1.x; cntx[5] = (float)x1.y; cntx[6] = (float)x1.z; cntx[7] = (float)x1.w;
    cntc[0] = (float)c0.x; cntc[1] = (float)c0.y; cntc[2] = (float)c0.z; cntc[3] = (float)c0.w;
    cntc[4] = (float)c1.x; cntc[5] = (float)c1.y; cntc[6] = (float)c1.z; cntc[7] = (float)c1.w;
  }

  v8f agg[4];

  // ---- self loop: relu(x @ Wxx + bxx) ----
  {
    v2f a[16];
#pragma unroll
    for (int kc = 0; kc < 16; ++kc) {
      a[kc].x = rowSelf[kc * 4 + kb];
      a[kc].y = rowSelf[kc * 4 + kb + 1];
    }
#pragma unroll
    for (int nt = 0; nt < 4; ++nt) {
      v8f acc = {};
#pragma unroll
      for (int kc = 0; kc < 16; ++kc)
        acc = wmma4(a[kc], bfrag(sWxx, kc, nt, lane), acc);
      const float bias = sBias[nt * 16 + mr];
#pragma unroll
      for (int r = 0; r < 8; ++r) agg[nt][r] = fmaxf(acc[r] + bias, 0.0f);
    }
  }

  // ---- x<->x edge message: count * relu(x[src_last] @ Wxex + bxex) ----
  {
    v2f a[16];
#pragma unroll
    for (int kc = 0; kc < 16; ++kc) {
      a[kc].x = rowGx[kc * 4 + kb];
      a[kc].y = rowGx[kc * 4 + kb + 1];
    }
#pragma unroll
    for (int nt = 0; nt < 4; ++nt) {
      v8f acc = {};
#pragma unroll
      for (int kc = 0; kc < 16; ++kc)
        acc = wmma4(a[kc], bfrag(sWxex, kc, nt, lane), acc);
      const float bias = sBias[64 + nt * 16 + mr];
#pragma unroll
      for (int r = 0; r < 8; ++r)
        agg[nt][r] += fmaxf(acc[r] + bias, 0.0f) * cntx[r];
    }
  }

  // ---- c->x edge message (K = 32): count * relu(c[src_last] @ Wcx + bcx) ----
  {
    v2f a[8];
#pragma unroll
    for (int kc = 0; kc < 8; ++kc) {
      a[kc].x = rowGc[kc * 4 + kb];
      a[kc].y = rowGc[kc * 4 + kb + 1];
    }
#pragma unroll
    for (int nt = 0; nt < 4; ++nt) {
      v8f acc = {};
#pragma unroll
      for (int kc = 0; kc < 8; ++kc)
        acc = wmma4(a[kc], bfrag(sWcx, kc, nt, lane), acc);
      const float bias = sBias[128 + nt * 16 + mr];
#pragma unroll
      for (int r = 0; r < 8; ++r)
        agg[nt][r] += fmaxf(acc[r] + bias, 0.0f) * cntc[r];
    }
  }

  // ---- pool: stage agg (C layout) to LDS, reload as A fragments ----
  float* st = sStage[wave];
#pragma unroll
  for (int nt = 0; nt < 4; ++nt)
#pragma unroll
    for (int r = 0; r < 8; ++r)
      st[(hb * 8 + r) * 64 + nt * 16 + mr] = agg[nt][r];
  __syncthreads();

  v2f pa[16];
#pragma unroll
  for (int kc = 0; kc < 16; ++kc)
    pa[kc] = *reinterpret_cast<const v2f*>(&st[mr * 64 + kc * 4 + kb]);

#pragma unroll
  for (int nt = 0; nt < 4; ++nt) {
    v8f acc = {};
#pragma unroll
    for (int kc = 0; kc < 16; ++kc)
      acc = wmma4(pa[kc], bfrag(sWpool, kc, nt, lane), acc);
    const float bias = sBias[192 + nt * 16 + mr];
    if (tvalid) {  // wave-uniform predicate: EXEC stays all-ones around WMMA
#pragma unroll
      for (int r = 0; r < 8; ++r)
        out[(size_t)(tbase + hb * 8 + r) * 64 + nt * 16 + mr] = acc[r] + bias;
    }
  }
}

// c_out = relu(c @ Wcc + bcc); K = 32
__global__ void __launch_bounds__(128)
c_forward_kernel(const float* __restrict__ cfeat, const float* __restrict__ Wcc,
                 const float* __restrict__ bcc, float* __restrict__ out) {
  __shared__ __align__(16) float sW[32 * 64];
  __shared__ __align__(16) float sB[64];
  const int tid = threadIdx.x;
  stage_swizzled(sW, Wcc, 32 * 64, tid);
  if (tid < 64) sB[tid] = bcc[tid];
  __syncthreads();

  const int wave = tid >> 5;
  const int lane = tid & 31;
  const int hb = lane >> 4;
  const int mr = lane & 15;
  const int kb = hb * 2;

  const int tile = blockIdx.x * 4 + wave;
  if (tile * 16 >= NCN) return;  // after the only barrier
  const int tbase = tile * 16;

  const float* rowS = cfeat + (size_t)(tbase + mr) * 32;
  v2f a[8];
#pragma unroll
  for (int kc = 0; kc < 8; ++kc) {
    a[kc].x = rowS[kc * 4 + kb];
    a[kc].y = rowS[kc * 4 + kb + 1];
  }

#pragma unroll
  for (int nt = 0; nt < 4; ++nt) {
    v8f acc = {};
#pragma unroll
    for (int kc = 0; kc < 8; ++kc)
      acc = wmma4(a[kc], bfrag(sW, kc, nt, lane), acc);
    const float bias = sB[nt * 16 + mr];
#pragma unroll
    for (int r = 0; r < 8; ++r)
      out[(size_t)(tbase + hb * 8 + r) * 64 + nt * 16 + mr] =
          fmaxf(acc[r] + bias, 0.0f);
  }
}

extern "C" void kernel_launch(void* const* d_in, const int* in_sizes, int n_in,
                              void* d_out, int out_size, void* d_ws, size_t ws_size,
                              hipStream_t stream) {
  (void)in_sizes; (void)n_in; (void)out_size; (void)ws_size;

  const float* x = (const float*)d_in[0];        // [100000, 64]
  const float* c = (const float*)d_in[1];        // [100000, 32]
  const int* edge_xx = (const int*)d_in[2];      // [2, 1600000]: src row 0, dst row 1
  const int* edge_cx = (const int*)d_in[3];      // [2, 800000]
  const float* Wxx = (const float*)d_in[4];
  const float* bxx = (const float*)d_in[5];
  const float* Wcc = (const float*)d_in[6];
  const float* bcc = (const float*)d_in[7];
  const float* Wxex = (const float*)d_in[8];
  const float* bxex = (const float*)d_in[9];
  const float* Wcx = (const float*)d_in[10];
  const float* bcx = (const float*)d_in[11];
  const float* Wpool = (const float*)d_in[12];
  const float* bpool = (const float*)d_in[13];
  float* out = (float*)d_out;  // x_out [100000*64] then c_out [100000*64]

  int* ws = (int*)d_ws;
  int* last_xx = ws;
  int* cnt_xx = ws + NXN;
  int* last_cx = ws + 2 * NXN;
  int* cnt_cx = ws + 3 * NXN;

  init_ws_kernel<<<(NXN + 255) / 256, 256, 0, stream>>>(last_xx, cnt_xx,
                                                        last_cx, cnt_cx);
  scatter_edges_kernel<<<(EXXE + 255) / 256, 256, 0, stream>>>(
      edge_xx + EXXE, EXXE, last_xx, cnt_xx);
  scatter_edges_kernel<<<(ECXE + 255) / 256, 256, 0, stream>>>(
      edge_cx + ECXE, ECXE, last_cx, cnt_cx);

  const int xTiles = NXN / 16;                 // 6250
  const int xBlocks = (xTiles + 3) / 4;        // 1563
  x_forward_kernel<<<xBlocks, 128, 0, stream>>>(
      x, c, edge_xx, edge_cx, last_xx, cnt_xx, last_cx, cnt_cx,
      Wxx, bxx, Wxex, bxex, Wcx, bcx, Wpool, bpool, out);

  const int cTiles = NCN / 16;                 // 6250
  const int cBlocks = (cTiles + 3) / 4;        // 1563
  c_forward_kernel<<<cBlocks, 128, 0, stream>>>(c, Wcc, bcc,
                                                out + (size_t)NXN * 64);
}